// ResFeature_18330920419811
// MI455X (gfx1250) — compile-verified
//
#include <hip/hip_runtime.h>
#include <hip/hip_bf16.h>
#include <float.h>

#define B_    4
#define L_    3000
#define K_    30
#define NA    6
#define ADIST 6
#define AANG  5
#define SEPS_ 1e-6f
#define LEPS_ 1000000.0f
#define PF_   4        // j-tile prefetch distance

typedef __attribute__((ext_vector_type(2))) float v2f;
typedef __attribute__((ext_vector_type(8))) float v8f;

// ---------------------------------------------------------------------------
// Kernel 1: per-residue features (15 dists + 3 angle cosines), residue-average
// coordinates and their squared norms (for the WMMA Gram trick).
// ---------------------------------------------------------------------------
__global__ void feat_avg_kernel(const float* __restrict__ coords,
                                const int*   __restrict__ mask,
                                float* __restrict__ feat_out,
                                float* __restrict__ avg,
                                float* __restrict__ norm2) {
    int idx = blockIdx.x * blockDim.x + threadIdx.x;
    if (idx >= B_ * L_) return;

    const float* c = coords + (size_t)idx * NA * 3;
    float x[NA], y[NA], z[NA];
    float ax = 0.f, ay = 0.f, az = 0.f;
#pragma unroll
    for (int a = 0; a < NA; ++a) {
        x[a] = c[a * 3 + 0];
        y[a] = c[a * 3 + 1];
        z[a] = c[a * 3 + 2];
        ax += x[a]; ay += y[a]; az += z[a];
    }
    const float inv = 1.0f / (float)NA;
    ax *= inv; ay *= inv; az *= inv;
    avg[(size_t)idx * 3 + 0] = ax;
    avg[(size_t)idx * 3 + 1] = ay;
    avg[(size_t)idx * 3 + 2] = az;
    norm2[idx] = ax * ax + ay * ay + az * az;

    const int mk = mask[idx];
    float* f = feat_out + (size_t)idx * 18;

    // 15 pairwise distances among first 6 atoms, triu(k=1) row-major order
    int p = 0;
#pragma unroll
    for (int i = 0; i < ADIST; ++i) {
#pragma unroll
        for (int j = i + 1; j < ADIST; ++j) {
            float dx = x[j] - x[i];
            float dy = y[j] - y[i];
            float dz = z[j] - z[i];
            float d  = sqrtf(dx * dx + dy * dy + dz * dz + SEPS_);
            f[p++] = mk ? d : LEPS_;
        }
    }

    // 3 angle cosines among first 5 atoms
    float vx[AANG - 1], vy[AANG - 1], vz[AANG - 1], nn[AANG - 1];
#pragma unroll
    for (int a = 0; a < AANG - 1; ++a) {
        vx[a] = x[a + 1] - x[a];
        vy[a] = y[a + 1] - y[a];
        vz[a] = z[a + 1] - z[a];
        nn[a] = sqrtf(vx[a] * vx[a] + vy[a] * vy[a] + vz[a] * vz[a]);
    }
#pragma unroll
    for (int a = 0; a < AANG - 2; ++a) {
        float dot = vx[a] * vx[a + 1] + vy[a] * vy[a + 1] + vz[a] * vz[a + 1];
        float ang = dot / (nn[a] * nn[a + 1] + SEPS_);
        f[15 + a] = mk ? ang : 0.0f;
    }
}

// ---------------------------------------------------------------------------
// Kernel 2: per-batch mask population (for the reference's pad_mask logic).
// ---------------------------------------------------------------------------
__global__ void masksum_kernel(const int* __restrict__ mask,
                               float* __restrict__ msum) {
    __shared__ float red[256];
    float s = 0.f;
    for (int l = threadIdx.x; l < L_; l += 256)
        s += (float)mask[blockIdx.x * L_ + l];
    red[threadIdx.x] = s;
    __syncthreads();
    for (int off = 128; off > 0; off >>= 1) {
        if (threadIdx.x < off) red[threadIdx.x] += red[threadIdx.x + off];
        __syncthreads();
    }
    if (threadIdx.x == 0) msum[blockIdx.x] = red[0];
}

// ---------------------------------------------------------------------------
// Kernel 3: kNN graph via V_WMMA_F32_16X16X4_F32 Gram tiles.
// One wave (32 threads) per 16-row i-tile; sweep 16-column j-tiles.
//   d^2[i][j] = |xi|^2 + |xj|^2 - 2 * (xi . xj),   K-dim 3 padded to 4.
// B-panel is L2-resident; global_prefetch_b8 hides the per-tile latency.
// ---------------------------------------------------------------------------
__global__ void __launch_bounds__(32)
knn_wmma_kernel(const float* __restrict__ avg,
                const float* __restrict__ norm2,
                const int*   __restrict__ mask,
                const float* __restrict__ msum,
                float* __restrict__ edge_out) {
    const int t  = threadIdx.x;   // 0..31
    const int b  = blockIdx.y;
    const int i0 = blockIdx.x * 16;

    __shared__ float tile[16][17];     // padded vs bank conflicts
    __shared__ float nI[16];
    __shared__ int   mI[16];
    __shared__ float kval[16][K_];
    __shared__ int   kidx[16][K_];

    for (int q = t; q < 16 * K_; q += 32) {
        ((float*)kval)[q] = FLT_MAX;
        ((int*)kidx)[q]   = -1;
    }

    const int lane_m = t & 15;
    const int hi     = t >> 4;
    const int kb     = hi * 2;         // K offset held by this half-wave

    if (t < 16) {
        int i  = i0 + t;
        int iC = (i < L_) ? i : (L_ - 1);
        nI[t] = norm2[b * L_ + iC];
        mI[t] = (i < L_) ? mask[b * L_ + i] : 0;
    }

    // A-matrix fragment: lane holds row (i0+lane_m), K = kb, kb+1 (z pad = 0)
    int iC = i0 + lane_m; if (iC >= L_) iC = L_ - 1;
    const float* arow = avg + (size_t)(b * L_ + iC) * 3;
    v2f A;
    A.x = arow[kb];
    A.y = (kb + 1 < 3) ? arow[kb + 1] : 0.0f;

    __syncthreads();

    const int NT = (L_ + 15) / 16;
    for (int jt = 0; jt < NT; ++jt) {
        const int j0 = jt * 16;

        // Prefetch the B-panel PF_ tiles ahead (one row per lane covers a
        // whole future 16-column tile) -> global_prefetch_b8.
        {
            int jp = j0 + PF_ * 16 + lane_m;
            if (jp < L_) {
                __builtin_prefetch(avg + (size_t)(b * L_ + jp) * 3, 0, 1);
                __builtin_prefetch(norm2 + (size_t)(b * L_ + jp), 0, 1);
            }
        }

        // B-matrix fragment: lane holds column (j0+lane_m), K = kb, kb+1
        int jB = j0 + lane_m;
        int jC = (jB < L_) ? jB : (L_ - 1);
        const float* brow = avg + (size_t)(b * L_ + jC) * 3;
        v2f Bm;
        Bm.x = brow[kb];
        Bm.y = (kb + 1 < 3) ? brow[kb + 1] : 0.0f;
        const float nJ = norm2[b * L_ + jC];
        const int   mJ = (jB < L_) ? mask[b * L_ + jC] : 0;

        v8f c = {};
        // D = A(16x4) x B(4x16) + 0  -> Gram tile of avg coordinates
        c = __builtin_amdgcn_wmma_f32_16x16x4_f32(
                /*neg_a=*/false, A, /*neg_b=*/false, Bm,
                /*c_mod=*/(short)0, c, /*reuse_a=*/false, /*reuse_b=*/false);

        const int jcol = j0 + lane_m;  // C layout: this lane's N == lane_m
#pragma unroll
        for (int r = 0; r < 8; ++r) {
            int   m  = r + 8 * hi;     // C layout: M = r + 8*(lane>=16)
            int   i  = i0 + m;
            float d2 = nI[m] + nJ - 2.0f * c[r];
            d2 = d2 > 0.0f ? d2 : 0.0f;
            float dist = sqrtf(d2 + SEPS_);
            float mm = (float)(mI[m] * mJ);
            dist = dist * mm + (1.0f - mm) * LEPS_;   // mask semantics
            if (i == jcol) dist += LEPS_;             // + eye * LEPS
            if (jcol >= L_ || i >= L_) dist = FLT_MAX;
            tile[m][lane_m] = dist;
        }
        __syncthreads();

        // Stable insertion into per-row sorted top-K (matches top_k tie order)
        if (t < 16 && (i0 + t) < L_) {
#pragma unroll 1
            for (int n = 0; n < 16; ++n) {
                int j = j0 + n;
                if (j >= L_) break;
                float cand = tile[t][n];
                if (cand < kval[t][K_ - 1]) {
                    int pos = K_ - 1;
                    while (pos > 0 && cand < kval[t][pos - 1]) {
                        kval[t][pos] = kval[t][pos - 1];
                        kidx[t][pos] = kidx[t][pos - 1];
                        --pos;
                    }
                    kval[t][pos] = cand;
                    kidx[t][pos] = j;
                }
            }
        }
        __syncthreads();
    }

    // Emit edges with self/pad/mask fix-ups per the reference
    if (t < 16) {
        int i = i0 + t;
        if (i < L_) {
            int   mk     = mI[t];
            float validf = mk ? (msum[b] - 1.0f) : 0.0f;
            if (validf < 0.0f) validf = 0.0f;
            float* eo = edge_out + (size_t)(b * L_ + i) * K_;
#pragma unroll 1
            for (int k = 0; k < K_; ++k) {
                int e = kidx[t][k];
                if (e == i) e = -1;
                if (validf < (float)k) e = -1;
                if (!mk) e = -1;
                eo[k] = (float)e;
            }
        }
    }
}

// ---------------------------------------------------------------------------
extern "C" void kernel_launch(void* const* d_in, const int* in_sizes, int n_in,
                              void* d_out, int out_size, void* d_ws, size_t ws_size,
                              hipStream_t stream) {
    (void)in_sizes; (void)n_in; (void)out_size; (void)ws_size;

    const float* coords = (const float*)d_in[0];   // (4,3000,6,3) f32
    const int*   mask   = (const int*)d_in[1];     // (4,3000) i32

    float* out  = (float*)d_out;
    float* feat = out;                                   // B*L*18
    float* edge = out + (size_t)B_ * L_ * 18;            // B*L*K (as f32)

    float* avg   = (float*)d_ws;                         // B*L*3
    float* norm2 = avg + (size_t)B_ * L_ * 3;            // B*L
    float* msum  = norm2 + (size_t)B_ * L_;              // B

    const int nres = B_ * L_;
    hipLaunchKernelGGL(feat_avg_kernel, dim3((nres + 255) / 256), dim3(256), 0,
                       stream, coords, mask, feat, avg, norm2);
    hipLaunchKernelGGL(masksum_kernel, dim3(B_), dim3(256), 0, stream,
                       mask, msum);
    dim3 g((L_ + 15) / 16, B_);
    hipLaunchKernelGGL(knn_wmma_kernel, g, dim3(32), 0, stream,
                       avg, norm2, mask, msum, edge);
}